// GraphNet_2911987827306
// MI455X (gfx1250) — compile-verified
//
#include <hip/hip_runtime.h>
#include <math.h>

#define N_NODES 50000
#define N_EDGES 800000
#define F_IN    512
#define H1      4
#define O_CH    64
#define HID     256   // H1*O_CH
#define SLOPE   0.01f
#define BN_EPS  1e-5f

typedef float v2f __attribute__((ext_vector_type(2)));
typedef float v8f __attribute__((ext_vector_type(8)));

// ---- monotone float<->uint encoding for atomicMax-based segment max ----
__device__ __forceinline__ unsigned enc_f(float f) {
    unsigned b = __float_as_uint(f);
    return (b & 0x80000000u) ? ~b : (b | 0x80000000u);
}
__device__ __forceinline__ float dec_f(unsigned u) {
    return (u & 0x80000000u) ? __uint_as_float(u ^ 0x80000000u)
                             : __uint_as_float(~u);
}

// ================= BatchNorm statistics =================
__global__ void col_stats_kernel(const float* __restrict__ x,
                                 float* __restrict__ csum, float* __restrict__ csq,
                                 int nrows, int ncols) {
    int col = blockIdx.x * blockDim.x + threadIdx.x;
    if (col >= ncols) return;
    int chunk = (nrows + gridDim.y - 1) / gridDim.y;
    int r0 = blockIdx.y * chunk;
    int r1 = min(r0 + chunk, nrows);
    float s = 0.f, s2 = 0.f;
    for (int r = r0; r < r1; ++r) {
        float v = x[(size_t)r * ncols + col];
        s += v; s2 += v * v;
    }
    atomicAdd(&csum[col], s);
    atomicAdd(&csq[col], s2);
}

// ReLU in place on [nrows, HID] + column stats for BN2 (fused pass over acc1).
__global__ void relu_stats_kernel(float* __restrict__ x,
                                  float* __restrict__ csum, float* __restrict__ csq,
                                  int nrows) {
    int col = threadIdx.x;                       // blockDim.x == HID == 256
    int chunk = (nrows + gridDim.x - 1) / gridDim.x;
    int r0 = blockIdx.x * chunk;
    int r1 = min(r0 + chunk, nrows);
    float s = 0.f, s2 = 0.f;
    for (int r = r0; r < r1; ++r) {
        size_t idx = (size_t)r * HID + col;
        float v = x[idx];
        v = v > 0.f ? v : 0.f;
        x[idx] = v;
        s += v; s2 += v * v;
    }
    atomicAdd(&csum[col], s);
    atomicAdd(&csq[col], s2);
}

// scale = gamma * rsqrt(var+eps); bias = beta - mu*scale  (BN folded to affine)
__global__ void finalize_bn_kernel(const float* __restrict__ csum, const float* __restrict__ csq,
                                   const float* __restrict__ gamma, const float* __restrict__ beta,
                                   float* __restrict__ scale, float* __restrict__ bias,
                                   int nrows, int ncols) {
    int c = blockIdx.x * blockDim.x + threadIdx.x;
    if (c >= ncols) return;
    float inv_n = 1.0f / (float)nrows;
    float mu  = csum[c] * inv_n;
    float var = csq[c] * inv_n - mu * mu;
    float sc  = gamma[c] * rsqrtf(var + BN_EPS);
    scale[c] = sc;
    bias[c]  = beta[c] - mu * sc;
}

// ================= weight prep: fold BN + pack B into WMMA-fragment layout =====
// Packed layout: pack[((k4*nCB + cb)*32 + lane)*2 + t] = B[k4*4 + koff(lane) + t][cb*16 + (lane&15)]
// where koff = (lane<16)?0:2. Each wave B-fragment then = one coalesced b64 load.
//
// Layer1 source: B[f, j] = scale1[f] * W1[j/64, f, j%64]     (K=512, NC=256)
__global__ void prep_pack1_kernel(const float* __restrict__ W1, const float* __restrict__ scale,
                                  float* __restrict__ pack) {
    const int nCB = HID >> 4; // 16
    int idx = blockIdx.x * blockDim.x + threadIdx.x;
    if (idx >= F_IN * HID) return;
    int t    = idx & 1;
    int lane = (idx >> 1) & 31;
    int rest = idx >> 6;
    int cb   = rest % nCB;
    int k4   = rest / nCB;
    int f = k4 * 4 + ((lane < 16) ? 0 : 2) + t;
    int j = cb * 16 + (lane & 15);
    int h = j >> 6, o = j & 63;
    pack[idx] = scale[f] * W1[((size_t)h * F_IN + f) * O_CH + o];
}
// Layer2 source: B[f, j] = scale2[f] * (j<64 ? W2[f,j] : Wres[f,j-64])   (K=256, NC=128)
__global__ void prep_pack2_kernel(const float* __restrict__ W2, const float* __restrict__ Wres,
                                  const float* __restrict__ scale2, float* __restrict__ pack) {
    const int nCB = 128 >> 4; // 8
    int idx = blockIdx.x * blockDim.x + threadIdx.x;
    if (idx >= HID * 128) return;
    int t    = idx & 1;
    int lane = (idx >> 1) & 31;
    int rest = idx >> 6;
    int cb   = rest % nCB;
    int k4   = rest / nCB;
    int f = k4 * 4 + ((lane < 16) ? 0 : 2) + t;
    int j = cb * 16 + (lane & 15);
    float v = (j < 64) ? W2[(size_t)f * O_CH + j] : Wres[(size_t)f * O_CH + (j - 64)];
    pack[idx] = scale2[f] * v;
}

// cvec1[h*64+o] = sum_f bias1[f] * W1[h,f,o]
__global__ void prep_c1_kernel(const float* __restrict__ W1, const float* __restrict__ bias,
                               float* __restrict__ cvec) {
    int j = blockIdx.x * blockDim.x + threadIdx.x;
    if (j >= HID) return;
    int h = j >> 6, o = j & 63;
    float s = 0.f;
    for (int f = 0; f < F_IN; ++f)
        s += bias[f] * W1[((size_t)h * F_IN + f) * O_CH + o];
    cvec[j] = s;
}
// cvec2[j]: j<64 -> sum_f bias2[f]*W2[f,j] ; j>=64 -> bres[j-64]+sum_f bias2[f]*Wres[f,j-64]
__global__ void prep_c2_kernel(const float* __restrict__ W2, const float* __restrict__ Wres,
                               const float* __restrict__ bres, const float* __restrict__ bias2,
                               float* __restrict__ cvec2) {
    int j = blockIdx.x * blockDim.x + threadIdx.x;
    if (j >= 128) return;
    float s = (j < 64) ? 0.f : bres[j - 64];
    const float* Wc = (j < 64) ? (W2 + j) : (Wres + (j - 64));
    for (int f = 0; f < HID; ++f)
        s += bias2[f] * Wc[(size_t)f * O_CH];
    cvec2[j] = s;
}

// ================= f32 WMMA GEMM =================
// C[nrows, NC] = A[nrows, K] * Bpack + biasvec[col].  B is fragment-packed (see above).
// Wave tile 32x64: 2 A-frags x 4 B-frags -> 8 v8f accumulators, 8 WMMAs per 6 b64 loads.
// Block = 8 waves -> 256x64 tile. Cols < split -> out0 (width split), else out1.
__global__ __launch_bounds__(256)
void wmma_gemm_kernel(const float* __restrict__ A, const float* __restrict__ Bpack,
                      const float* __restrict__ biasvec,
                      float* __restrict__ out0, float* __restrict__ out1,
                      int nrows, int K, int NC, int split) {
    const int wave = threadIdx.x >> 5;
    const int lane = threadIdx.x & 31;
    const int row0 = (blockIdx.x * 8 + wave) * 32;
    const int col0 = blockIdx.y * 64;
    const int nCB  = NC >> 4;

    int m0 = row0 + (lane & 15);
    int m1 = m0 + 16;
    int m0c = m0 < nrows ? m0 : nrows - 1;        // clamp: keep EXEC full for WMMA
    int m1c = m1 < nrows ? m1 : nrows - 1;
    const int koff = (lane < 16) ? 0 : 2;         // f32 A frag: V0=K{0|2}, V1=K{1|3}

    v8f acc[2][4] = {};
    const float* A0 = A + (size_t)m0c * K + koff;
    const float* A1 = A + (size_t)m1c * K + koff;
    // wave's packed-B base: 4 consecutive col-blocks starting at blockIdx.y*4
    const float* Bbase = Bpack + ((size_t)blockIdx.y * 4) * 64 + lane * 2;
    const size_t Bstride = (size_t)nCB * 64;      // floats per k-quad

    for (int k0 = 0; k0 < K; k0 += 4) {
        v2f a0 = *(const v2f*)(A0 + k0);
        v2f a1 = *(const v2f*)(A1 + k0);
        const float* Bp = Bbase + (size_t)(k0 >> 2) * Bstride;
        v2f b0 = *(const v2f*)(Bp);
        v2f b1 = *(const v2f*)(Bp + 64);
        v2f b2 = *(const v2f*)(Bp + 128);
        v2f b3 = *(const v2f*)(Bp + 192);
        acc[0][0] = __builtin_amdgcn_wmma_f32_16x16x4_f32(false, a0, false, b0, (short)0, acc[0][0], false, false);
        acc[0][1] = __builtin_amdgcn_wmma_f32_16x16x4_f32(false, a0, false, b1, (short)0, acc[0][1], false, false);
        acc[0][2] = __builtin_amdgcn_wmma_f32_16x16x4_f32(false, a0, false, b2, (short)0, acc[0][2], false, false);
        acc[0][3] = __builtin_amdgcn_wmma_f32_16x16x4_f32(false, a0, false, b3, (short)0, acc[0][3], false, false);
        acc[1][0] = __builtin_amdgcn_wmma_f32_16x16x4_f32(false, a1, false, b0, (short)0, acc[1][0], false, false);
        acc[1][1] = __builtin_amdgcn_wmma_f32_16x16x4_f32(false, a1, false, b1, (short)0, acc[1][1], false, false);
        acc[1][2] = __builtin_amdgcn_wmma_f32_16x16x4_f32(false, a1, false, b2, (short)0, acc[1][2], false, false);
        acc[1][3] = __builtin_amdgcn_wmma_f32_16x16x4_f32(false, a1, false, b3, (short)0, acc[1][3], false, false);
    }

    // C/D layout: lane<16 -> n=lane, m=+v ; lane>=16 -> n=lane-16, m=+8+v
    const int nsub = lane & 15;
    const int rsel = (lane < 16) ? 0 : 8;
    #pragma unroll
    for (int rt = 0; rt < 2; ++rt) {
        int rbase = row0 + rt * 16 + rsel;
        #pragma unroll
        for (int c = 0; c < 4; ++c) {
            int col = col0 + c * 16 + nsub;
            float bv = biasvec[col];
            #pragma unroll
            for (int v = 0; v < 8; ++v) {
                int r = rbase + v;
                if (r < nrows) {
                    float val = acc[rt][c][v] + bv;
                    if (col < split) out0[(size_t)r * split + col] = val;
                    else             out1[(size_t)r * (NC - split) + (col - split)] = val;
                }
            }
        }
    }
}

// ================= attention scalar terms: a1[n,h]=ft·al, a2[n,h]=ft·ar =================
__global__ void att_kernel(const float* __restrict__ ft, const float* __restrict__ al,
                           const float* __restrict__ ar, float* __restrict__ a1,
                           float* __restrict__ a2, int nnodes, int H) {
    int wv = blockIdx.x * (blockDim.x >> 5) + (threadIdx.x >> 5);
    int lane = threadIdx.x & 31;
    int nw = gridDim.x * (blockDim.x >> 5);
    int total = nnodes * H;
    for (int p = wv; p < total; p += nw) {
        int n = p / H, h = p - n * H;
        const float* f = ft + (size_t)n * (H * 64) + h * 64;
        const float* alh = al + h * 64;
        const float* arh = ar + h * 64;
        float s1 = f[lane] * alh[lane] + f[lane + 32] * alh[lane + 32];
        float s2 = f[lane] * arh[lane] + f[lane + 32] * arh[lane + 32];
        for (int off = 16; off; off >>= 1) {
            s1 += __shfl_xor(s1, off, 32);
            s2 += __shfl_xor(s2, off, 32);
        }
        if (lane == 0) { a1[p] = s1; a2[p] = s2; }
    }
}

// ================= edge-phase kernels (segment softmax + aggregation) =================
__global__ void edge_max_kernel(const int* __restrict__ ei, const float* __restrict__ a1,
                                const float* __restrict__ a2, unsigned* __restrict__ smax,
                                int ne, int H) {
    int total = ne * H;
    for (int t = blockIdx.x * blockDim.x + threadIdx.x; t < total;
         t += gridDim.x * blockDim.x) {
        int e = t / H, h = t - e * H;
        int s_ = ei[e], d_ = ei[ne + e];
        float s = a1[d_ * H + h] + a2[s_ * H + h];
        s = s > 0.f ? s : s * SLOPE;
        atomicMax(&smax[d_ * H + h], enc_f(s));
    }
}

__global__ void edge_exp_kernel(const int* __restrict__ ei, const float* __restrict__ a1,
                                const float* __restrict__ a2, const unsigned* __restrict__ smax,
                                float* __restrict__ ew, float* __restrict__ denom,
                                int ne, int H) {
    int total = ne * H;
    for (int t = blockIdx.x * blockDim.x + threadIdx.x; t < total;
         t += gridDim.x * blockDim.x) {
        int e = t / H, h = t - e * H;
        int s_ = ei[e], d_ = ei[ne + e];
        float s = a1[d_ * H + h] + a2[s_ * H + h];
        s = s > 0.f ? s : s * SLOPE;
        float v = expf(s - dec_f(smax[d_ * H + h]));
        ew[t] = v;
        atomicAdd(&denom[d_ * H + h], v);
    }
}

// One wave per edge; lanes cover H*64 feature channels; scatter via f32 atomics.
__global__ void edge_agg_kernel(const int* __restrict__ ei, const float* __restrict__ ft,
                                const float* __restrict__ ew, const float* __restrict__ denom,
                                float* __restrict__ acc, int ne, int H) {
    int wv = blockIdx.x * (blockDim.x >> 5) + (threadIdx.x >> 5);
    int lane = threadIdx.x & 31;
    int nw = gridDim.x * (blockDim.x >> 5);
    int W = H * 64;
    for (int e = wv; e < ne; e += nw) {
        int s_ = ei[e], d_ = ei[ne + e];
        const float* fs = ft + (size_t)s_ * W;
        float* ad = acc + (size_t)d_ * W;
        for (int i = lane; i < W; i += 32) {
            int h = i >> 6;
            float coeff = ew[(size_t)e * H + h] / denom[d_ * H + h];
            atomicAdd(&ad[i], coeff * fs[i]);
        }
    }
}

// ================= host launcher =================
extern "C" void kernel_launch(void* const* d_in, const int* in_sizes, int n_in,
                              void* d_out, int out_size, void* d_ws, size_t ws_size,
                              hipStream_t stream) {
    const float* features  = (const float*)d_in[0];
    const int*   ei        = (const int*)  d_in[1];
    const float* bn1_gamma = (const float*)d_in[2];
    const float* bn1_beta  = (const float*)d_in[3];
    const float* W1        = (const float*)d_in[4];
    const float* al1       = (const float*)d_in[5];
    const float* ar1       = (const float*)d_in[6];
    const float* bn2_gamma = (const float*)d_in[7];
    const float* bn2_beta  = (const float*)d_in[8];
    const float* W2        = (const float*)d_in[9];
    const float* al2       = (const float*)d_in[10];
    const float* ar2       = (const float*)d_in[11];
    const float* Wres      = (const float*)d_in[12];
    const float* bres      = (const float*)d_in[13];
    float* out = (float*)d_out;
    float* w   = (float*)d_ws;

    // ---- workspace layout (floats) ----
    size_t off = 0;
    float* ft1   = w + off; off += (size_t)N_NODES * HID;     // layer1 projections
    float* acc1  = w + off; off += (size_t)N_NODES * HID;     // layer1 aggregate -> h1
    float* ft2   = w + off; off += (size_t)N_NODES * O_CH;    // layer2 projections
    float* ew1   = w + off; off += (size_t)N_EDGES * H1;      // edge exp numerators
    float* ew2   = w + off; off += (size_t)N_EDGES;
    float* a1_1  = w + off; off += (size_t)N_NODES * H1;
    float* a2_1  = w + off; off += (size_t)N_NODES * H1;
    float* a1_2  = w + off; off += (size_t)N_NODES;
    float* a2_2  = w + off; off += (size_t)N_NODES;
    float* zstart = w + off;                                  // ---- zeroed block ----
    unsigned* smax1 = (unsigned*)(w + off); off += (size_t)N_NODES * H1;
    float* den1  = w + off; off += (size_t)N_NODES * H1;
    unsigned* smax2 = (unsigned*)(w + off); off += (size_t)N_NODES;
    float* den2  = w + off; off += (size_t)N_NODES;
    float* csum1 = w + off; off += F_IN;
    float* csq1  = w + off; off += F_IN;
    float* csum2 = w + off; off += HID;
    float* csq2  = w + off; off += HID;
    size_t zlen  = (size_t)((w + off) - zstart);              // ---- end zero block ----
    float* scale1 = w + off; off += F_IN;
    float* bias1  = w + off; off += F_IN;
    float* scale2 = w + off; off += HID;
    float* bias2  = w + off; off += HID;
    float* B1p    = w + off; off += (size_t)F_IN * HID;       // packed [512,256]
    float* cvec1  = w + off; off += HID;
    float* B2p    = w + off; off += (size_t)HID * 128;        // packed [256,128]
    float* cvec2  = w + off; off += 128;

    // per-call zero init (capture-safe)
    hipMemsetAsync(zstart, 0, zlen * sizeof(float), stream);
    hipMemsetAsync(acc1, 0, (size_t)N_NODES * HID * sizeof(float), stream);

    // ---- BN1 stats + fold into packed W1 panel ----
    col_stats_kernel<<<dim3(F_IN / 256, 128), 256, 0, stream>>>(features, csum1, csq1, N_NODES, F_IN);
    finalize_bn_kernel<<<2, 256, 0, stream>>>(csum1, csq1, bn1_gamma, bn1_beta, scale1, bias1, N_NODES, F_IN);
    prep_pack1_kernel<<<(F_IN * HID) / 256, 256, 0, stream>>>(W1, scale1, B1p);
    prep_c1_kernel<<<1, 256, 0, stream>>>(W1, bias1, cvec1);

    // ---- layer1 projection (WMMA f32, 32x64 wave tiles): ft1 = x @ W1' + c1 ----
    wmma_gemm_kernel<<<dim3((N_NODES + 255) / 256, HID / 64), 256, 0, stream>>>(
        features, B1p, cvec1, ft1, ft1, N_NODES, F_IN, HID, HID);

    // ---- layer1 attention + segment softmax + aggregation ----
    att_kernel<<<1024, 256, 0, stream>>>(ft1, al1, ar1, a1_1, a2_1, N_NODES, H1);
    edge_max_kernel<<<4096, 256, 0, stream>>>(ei, a1_1, a2_1, smax1, N_EDGES, H1);
    edge_exp_kernel<<<4096, 256, 0, stream>>>(ei, a1_1, a2_1, smax1, ew1, den1, N_EDGES, H1);
    edge_agg_kernel<<<8192, 256, 0, stream>>>(ei, ft1, ew1, den1, acc1, N_EDGES, H1);

    // ---- h1 = relu(acc1) in place + BN2 stats; fold BN2 into packed [W2|Wres] ----
    relu_stats_kernel<<<256, 256, 0, stream>>>(acc1, csum2, csq2, N_NODES);
    finalize_bn_kernel<<<1, 256, 0, stream>>>(csum2, csq2, bn2_gamma, bn2_beta, scale2, bias2, N_NODES, HID);
    prep_pack2_kernel<<<(HID * 128) / 256, 256, 0, stream>>>(W2, Wres, scale2, B2p);
    prep_c2_kernel<<<1, 128, 0, stream>>>(W2, Wres, bres, bias2, cvec2);

    // ---- layer2 projection + residual in one WMMA GEMM:
    //      cols 0..63 -> ft2, cols 64..127 -> d_out (= x2@Wres + bres)
    wmma_gemm_kernel<<<dim3((N_NODES + 255) / 256, 128 / 64), 256, 0, stream>>>(
        acc1, B2p, cvec2, ft2, out, N_NODES, HID, 128, 64);

    // ---- layer2 attention + softmax + aggregation (accumulates into d_out) ----
    att_kernel<<<512, 256, 0, stream>>>(ft2, al2, ar2, a1_2, a2_2, N_NODES, 1);
    edge_max_kernel<<<2048, 256, 0, stream>>>(ei, a1_2, a2_2, smax2, N_EDGES, 1);
    edge_exp_kernel<<<2048, 256, 0, stream>>>(ei, a1_2, a2_2, smax2, ew2, den2, N_EDGES, 1);
    edge_agg_kernel<<<8192, 256, 0, stream>>>(ei, ft2, ew2, den2, out, N_EDGES, 1);
}